// GuidedFilterND_6193342841708
// MI455X (gfx1250) — compile-verified
//
#include <hip/hip_runtime.h>

#define HH 768
#define WW 768
#define CC 4
#define NB 8
#define RAD 8
#define EPSV 1e-4f

typedef __attribute__((ext_vector_type(2))) float v2f;
typedef __attribute__((ext_vector_type(8))) float v8f;

static __device__ __forceinline__ v8f wmma4(v2f a, v2f b, v8f c) {
  // V_WMMA_F32_16X16X4_F32: D(16x16) = A(16x4) * B(4x16) + C
  return __builtin_amdgcn_wmma_f32_16x16x4_f32(false, a, false, b, (short)0, c,
                                               false, false);
}

// Radius-8, zero-padded 2D box filter of a 32x32 LDS tile (stride 33).
// Output: 16x16 tile in WMMA f32 C/D layout (reg v, lane l -> row v+8*(l>>4),
// col l&15). T is a 32x17 LDS transpose/intermediate buffer.
// band[kk] holds the 0/1 banded-matrix operand shared by both passes:
//   value(reg r) = 1 iff 0 <= (4*kk + 2*(lane>>4) + r) - (lane&15) <= 16
// Horizontal pass uses it as B (cols), vertical pass as A (rows).
// All 8 LDS operand pairs are fetched before each MMA chain so the ds_loads
// pipeline instead of serializing on a single reused register pair.
static __device__ __forceinline__ v8f box2d_16(const float* X, float* T,
                                               const v2f band[8], int lane) {
  const int m = lane & 15;   // row (A) / col (B,D) index for this lane
  const int hi = lane >> 4;  // lane half selects K offset (+2)
  #pragma unroll
  for (int rs = 0; rs < 2; ++rs) {  // intermediate rows 0..15 and 16..31
    v2f a[8];
    #pragma unroll
    for (int kk = 0; kk < 8; ++kk) {
      const int j0 = 4 * kk + 2 * hi;  // K for reg 0 (reg1 = +1)
      v2f t = { X[(rs * 16 + m) * 33 + j0],
                X[(rs * 16 + m) * 33 + j0 + 1] };
      a[kk] = t;
    }
    v8f acc = {};
    #pragma unroll
    for (int kk = 0; kk < 8; ++kk) acc = wmma4(a[kk], band[kk], acc);
    #pragma unroll
    for (int v = 0; v < 8; ++v)
      T[(rs * 16 + v + 8 * hi) * 17 + m] = acc[v];
  }
  __syncthreads();
  v2f b[8];
  #pragma unroll
  for (int kk = 0; kk < 8; ++kk) {
    const int j0 = 4 * kk + 2 * hi;
    v2f t = { T[j0 * 17 + m], T[(j0 + 1) * 17 + m] };
    b[kk] = t;
  }
  v8f acc = {};
  #pragma unroll
  for (int kk = 0; kk < 8; ++kk) acc = wmma4(band[kk], b[kk], acc);
  __syncthreads();
  return acc;
}

static __device__ __forceinline__ void make_band(v2f band[8], int lane) {
  const int hi = lane >> 4;
  const int m = lane & 15;
  #pragma unroll
  for (int kk = 0; kk < 8; ++kk) {
    const int d0 = 4 * kk + 2 * hi - m;
    v2f t = { (d0 >= 0 && d0 <= 16) ? 1.f : 0.f,
              (d0 + 1 >= 0 && d0 + 1 <= 16) ? 1.f : 0.f };
    band[kk] = t;
  }
}

// Row-sums of inverse(M) for symmetric 4x4 M (adjugate method).
static __device__ __forceinline__ void inv4_rowsum(const float m[16],
                                                   float rs[4]) {
  float i0  =  m[5]*m[10]*m[15] - m[5]*m[11]*m[14] - m[9]*m[6]*m[15] + m[9]*m[7]*m[14] + m[13]*m[6]*m[11] - m[13]*m[7]*m[10];
  float i4  = -m[4]*m[10]*m[15] + m[4]*m[11]*m[14] + m[8]*m[6]*m[15] - m[8]*m[7]*m[14] - m[12]*m[6]*m[11] + m[12]*m[7]*m[10];
  float i8  =  m[4]*m[9]*m[15] - m[4]*m[11]*m[13] - m[8]*m[5]*m[15] + m[8]*m[7]*m[13] + m[12]*m[5]*m[11] - m[12]*m[7]*m[9];
  float i12 = -m[4]*m[9]*m[14] + m[4]*m[10]*m[13] + m[8]*m[5]*m[14] - m[8]*m[6]*m[13] - m[12]*m[5]*m[10] + m[12]*m[6]*m[9];
  float i1  = -m[1]*m[10]*m[15] + m[1]*m[11]*m[14] + m[9]*m[2]*m[15] - m[9]*m[3]*m[14] - m[13]*m[2]*m[11] + m[13]*m[3]*m[10];
  float i5  =  m[0]*m[10]*m[15] - m[0]*m[11]*m[14] - m[8]*m[2]*m[15] + m[8]*m[3]*m[14] + m[12]*m[2]*m[11] - m[12]*m[3]*m[10];
  float i9  = -m[0]*m[9]*m[15] + m[0]*m[11]*m[13] + m[8]*m[1]*m[15] - m[8]*m[3]*m[13] - m[12]*m[1]*m[11] + m[12]*m[3]*m[9];
  float i13 =  m[0]*m[9]*m[14] - m[0]*m[10]*m[13] - m[8]*m[1]*m[14] + m[8]*m[2]*m[13] + m[12]*m[1]*m[10] - m[12]*m[2]*m[9];
  float i2  =  m[1]*m[6]*m[15] - m[1]*m[7]*m[14] - m[5]*m[2]*m[15] + m[5]*m[3]*m[14] + m[13]*m[2]*m[7] - m[13]*m[3]*m[6];
  float i6  = -m[0]*m[6]*m[15] + m[0]*m[7]*m[14] + m[4]*m[2]*m[15] - m[4]*m[3]*m[14] - m[12]*m[2]*m[7] + m[12]*m[3]*m[6];
  float i10 =  m[0]*m[5]*m[15] - m[0]*m[7]*m[13] - m[4]*m[1]*m[15] + m[4]*m[3]*m[13] + m[12]*m[1]*m[7] - m[12]*m[3]*m[5];
  float i14 = -m[0]*m[5]*m[14] + m[0]*m[6]*m[13] + m[4]*m[1]*m[14] - m[4]*m[2]*m[13] - m[12]*m[1]*m[6] + m[12]*m[2]*m[5];
  float i3  = -m[1]*m[6]*m[11] + m[1]*m[7]*m[10] + m[5]*m[2]*m[11] - m[5]*m[3]*m[10] - m[9]*m[2]*m[7] + m[9]*m[3]*m[6];
  float i7  =  m[0]*m[6]*m[11] - m[0]*m[7]*m[10] - m[4]*m[2]*m[11] + m[4]*m[3]*m[10] + m[8]*m[2]*m[7] - m[8]*m[3]*m[6];
  float i11 = -m[0]*m[5]*m[11] + m[0]*m[7]*m[9] + m[4]*m[1]*m[11] - m[4]*m[3]*m[9] - m[8]*m[1]*m[7] + m[8]*m[3]*m[5];
  float i15 =  m[0]*m[5]*m[10] - m[0]*m[6]*m[9] - m[4]*m[1]*m[10] + m[4]*m[2]*m[9] + m[8]*m[1]*m[6] - m[8]*m[2]*m[5];
  const float det = m[0]*i0 + m[1]*i4 + m[2]*i8 + m[3]*i12;
  const float id = 1.0f / det;
  rs[0] = (i0 + i1 + i2 + i3) * id;
  rs[1] = (i4 + i5 + i6 + i7) * id;
  rs[2] = (i8 + i9 + i10 + i11) * id;
  rs[3] = (i12 + i13 + i14 + i15) * id;
}

// Stage 1: box-filter 19 product planes (I, p, p*I, I⊗I upper-tri), then per
// pixel compute A (4ch) and b (1ch). One wave per 16x16 output tile.
__global__ __launch_bounds__(32) void gf_stats(const float* __restrict__ I,
                                               const float* __restrict__ P,
                                               float* __restrict__ Aout,
                                               float* __restrict__ Bout) {
  __shared__ float raw[5][32 * 33];  // I0..I3, p halo tiles
  __shared__ float X[32 * 33];       // product-plane staging
  __shared__ float T[32 * 17];       // horizontal-pass intermediate
  const int lane = threadIdx.x;
  const int tx = blockIdx.x * 16;
  const int ty = blockIdx.y * 16;
  const int bb = blockIdx.z;
  const int gc = tx - RAD + lane;    // lane = halo column
  const bool cok = (gc >= 0) && (gc < WW);
  const float* Ib = I + (size_t)bb * CC * HH * WW;
  const float* Pb = P + (size_t)bb * HH * WW;
  for (int y = 0; y < 32; ++y) {
    const int gr = ty - RAD + y;
    const bool ok = cok && (gr >= 0) && (gr < HH);
    const size_t off = (size_t)gr * WW + gc;
    raw[4][y * 33 + lane] = ok ? Pb[off] : 0.f;
    #pragma unroll
    for (int c = 0; c < CC; ++c)
      raw[c][y * 33 + lane] = ok ? Ib[(size_t)c * HH * WW + off] : 0.f;
  }
  __syncthreads();

  v2f band[8];
  make_band(band, lane);

  v8f S[19];
  // Planes 0..4 are raw I_c / p: filter the raw tiles in place (no staging).
  #pragma unroll
  for (int pl = 0; pl < 5; ++pl)
    S[pl] = box2d_16(&raw[pl][0], T, band, lane);

  // Planes 5..8 = p*I_c ; 9..18 = I_a*I_b (a<=b): stage product into X.
  constexpr int ia[14]  = {4,4,4,4, 0,0,0,0, 1,1,1, 2,2, 3};
  constexpr int ibx[14] = {0,1,2,3, 0,1,2,3, 1,2,3, 2,3, 3};
  #pragma unroll
  for (int pl = 0; pl < 14; ++pl) {
    #pragma unroll
    for (int y = 0; y < 32; ++y)
      X[y * 33 + lane] =
          raw[ia[pl]][y * 33 + lane] * raw[ibx[pl]][y * 33 + lane];
    __syncthreads();
    S[5 + pl] = box2d_16(X, T, band, lane);
  }

  // Per-pixel algebra; lane covers col tx+(lane&15), rows ty+v+8*(lane>>4).
  const int hi = lane >> 4;
  const int w = lane & 15;
  const int col = tx + w;
  const int cw = min(col + RAD, WW - 1) - max(col - RAD, 0) + 1;
  #pragma unroll
  for (int v = 0; v < 8; ++v) {
    const int row = ty + v + 8 * hi;
    const int chc = min(row + RAD, HH - 1) - max(row - RAD, 0) + 1;
    const float invN = 1.f / (float)(chc * cw);
    const float pm = S[4][v] * invN;
    float Im[4], first[4];
    float sumIm = 0.f;
    #pragma unroll
    for (int c = 0; c < 4; ++c) {
      Im[c] = S[c][v] * invN;
      sumIm += Im[c];
      first[c] = S[5 + c][v] * invN - pm * Im[c];
    }
    // Un-centered second-moment matrix (as in the reference) + eps*I.
    float M[16];
    M[0]  = S[9][v]  * invN + EPSV;
    M[1]  = M[4]  = S[10][v] * invN;
    M[2]  = M[8]  = S[11][v] * invN;
    M[3]  = M[12] = S[12][v] * invN;
    M[5]  = S[13][v] * invN + EPSV;
    M[6]  = M[9]  = S[14][v] * invN;
    M[7]  = M[13] = S[15][v] * invN;
    M[10] = S[16][v] * invN + EPSV;
    M[11] = M[14] = S[17][v] * invN;
    M[15] = S[18][v] * invN + EPSV;
    float rs[4];
    inv4_rowsum(M, rs);
    float sumA = 0.f;
    float Ac[4];
    #pragma unroll
    for (int c = 0; c < 4; ++c) {
      Ac[c] = first[c] * rs[c];   // einsum 'bchw,bhwcd->bchw'
      sumA += Ac[c];
    }
    const float bv = pm - sumA * sumIm;  // einsum 'bchw,bdhw->bhw'
    #pragma unroll
    for (int c = 0; c < 4; ++c)
      Aout[(((size_t)bb * CC + c) * HH + row) * WW + col] = Ac[c];
    Bout[((size_t)bb * HH + row) * WW + col] = bv;
  }
}

// Stage 2: box-filter A (4ch) and b, fuse q = (sum_c A_mean_c)*(sum_c I_c) + b_mean
__global__ __launch_bounds__(32) void gf_final(const float* __restrict__ I,
                                               const float* __restrict__ Ain,
                                               const float* __restrict__ Bin,
                                               float* __restrict__ Q) {
  __shared__ float X[32 * 33];
  __shared__ float T[32 * 17];
  const int lane = threadIdx.x;
  const int tx = blockIdx.x * 16;
  const int ty = blockIdx.y * 16;
  const int bb = blockIdx.z;
  const int gc = tx - RAD + lane;
  const bool cok = (gc >= 0) && (gc < WW);
  v2f band[8];
  make_band(band, lane);
  v8f S[5];
  #pragma unroll
  for (int pl = 0; pl < 5; ++pl) {
    const float* src = (pl < 4) ? (Ain + (((size_t)bb * CC + pl) * (size_t)HH) * WW)
                                : (Bin + (size_t)bb * HH * WW);
    for (int y = 0; y < 32; ++y) {
      const int gr = ty - RAD + y;
      const bool ok = cok && (gr >= 0) && (gr < HH);
      X[y * 33 + lane] = ok ? src[(size_t)gr * WW + gc] : 0.f;
    }
    __syncthreads();
    S[pl] = box2d_16(X, T, band, lane);
  }
  const int hi = lane >> 4;
  const int w = lane & 15;
  const int col = tx + w;
  const int cw = min(col + RAD, WW - 1) - max(col - RAD, 0) + 1;
  const float* Ib = I + (size_t)bb * CC * HH * WW;
  #pragma unroll
  for (int v = 0; v < 8; ++v) {
    const int row = ty + v + 8 * hi;
    const int chc = min(row + RAD, HH - 1) - max(row - RAD, 0) + 1;
    const float invN = 1.f / (float)(chc * cw);
    const size_t off = (size_t)row * WW + col;
    float sumI = 0.f;
    #pragma unroll
    for (int c = 0; c < 4; ++c) sumI += Ib[(size_t)c * HH * WW + off];
    const float am = (S[0][v] + S[1][v] + S[2][v] + S[3][v]) * invN;
    const float bm = S[4][v] * invN;
    Q[((size_t)bb * HH + row) * WW + col] = am * sumI + bm;
  }
}

extern "C" void kernel_launch(void* const* d_in, const int* in_sizes, int n_in,
                              void* d_out, int out_size, void* d_ws,
                              size_t ws_size, hipStream_t stream) {
  const float* I = (const float*)d_in[0];
  const float* P = (const float*)d_in[1];
  float* Q = (float*)d_out;
  float* Aws = (float*)d_ws;                               // 8*4*768*768 f32
  float* Bws = Aws + (size_t)NB * CC * HH * WW;            // 8*768*768 f32
  dim3 grid(WW / 16, HH / 16, NB);
  gf_stats<<<grid, dim3(32), 0, stream>>>(I, P, Aws, Bws);
  gf_final<<<grid, dim3(32), 0, stream>>>(I, Aws, Bws, Q);
  (void)in_sizes; (void)n_in; (void)out_size; (void)ws_size;
}